// TPS2RGBSplineXY_30124900614365
// MI455X (gfx1250) — compile-verified
//
#include <hip/hip_runtime.h>
#include <hip/hip_bf16.h>
#include <math.h>

// ---------------------------------------------------------------------------
// TPS RGB spline on MI455X (gfx1250, wave32).
//
// Phase 1 (tiny): per (batch, channel) solve the 36x36 TPS system with
//                 Gauss-Jordan + partial pivoting (replaces pinv; K is
//                 invertible for random knots with lambda > 0).
//                 Store weights padded to (36,16) fp32 so they can be used
//                 directly as WMMA B-operands (columns 3..15 are zero).
//
// Phase 2 (hot, memory-bound ~40MB traffic): each wave handles 16 pixels.
//   A-tile (16x36 features = distances to 32 knots, 1, r, g, b) is built in
//   registers chunk-by-chunk (K=4) and fed to V_WMMA_F32_16X16X4_F32,
//   accumulating a 16x16 f32 D-tile of which columns 0..2 are the output
//   channels. D is staged through LDS and stored as coalesced runs of
//   48 consecutive floats per wave (pixel-major, 3 floats per pixel, which
//   matches raw.reshape(B,HW,3) exactly).
//
//   sqrt: arguments are squared RGB distances in [0,3] (always normal
//   range), so use the raw v_sqrt_f32 (<=1 ulp) via __builtin_amdgcn_sqrtf
//   instead of libm's correctly-rounded sequence (saves ~8 VALU ops/sqrt).
//   sqrt(0)==0 exactly, so the reference's where(sq>0,...) guard (gradient
//   safety only) is dropped.
// ---------------------------------------------------------------------------

#define NK 32
#define NF 36              // NK + 4
#define HW (448 * 448)     // 200704
#define BATCH 8
#define PIX_PER_BATCH_ELEMS (3 * HW)   // 602112 floats per batch
#define PARAM_STRIDE (6 * NK + 3)      // 195
#define TILES_PER_BATCH (HW / 16)      // 12544
#define WAVES_PER_BLOCK 8
#define BLOCKS_PER_BATCH (TILES_PER_BATCH / WAVES_PER_BLOCK)  // 1568

typedef __attribute__((ext_vector_type(2))) float v2f;
typedef __attribute__((ext_vector_type(8))) float v8f;

// ---------------------------------------------------------------------------
// Kernel 1: build + solve the 36x36 augmented system per (b, channel).
// Grid: BATCH*3 blocks, 128 threads.
// wpad layout: wpad[b][row<36][col<16], col==ch holds weight, cols 3..15 == 0.
// ---------------------------------------------------------------------------
__global__ void tps_solve_kernel(const float* __restrict__ params,
                                 float* __restrict__ wpad) {
    const int b  = blockIdx.x / 3;
    const int ch = blockIdx.x % 3;
    const int tid = threadIdx.x;
    const int bd  = blockDim.x;
    const float* P = params + b * PARAM_STRIDE;

    __shared__ float K[NF][NF + 1];   // augmented: col 36 = rhs
    __shared__ float xs[NK][3];
    __shared__ float piv[NF + 1];
    __shared__ float fac[NF];
    __shared__ int   pivIdx;

    if (tid < NK * 3) xs[tid / 3][tid % 3] = P[tid];
    __syncthreads();

    const float lam = P[6 * NK + ch];

    for (int idx = tid; idx < NF * (NF + 1); idx += bd) {
        const int row = idx / (NF + 1);
        const int col = idx % (NF + 1);
        float val;
        if (col == NF) {
            val = (row < NK) ? P[3 * NK + row * 3 + ch] : 0.0f;   // rhs
        } else if (row < NK && col < NK) {
            const float dx = xs[row][0] - xs[col][0];
            const float dy = xs[row][1] - xs[col][1];
            const float dz = xs[row][2] - xs[col][2];
            const float sq = dx * dx + dy * dy + dz * dz;
            val = __builtin_amdgcn_sqrtf(sq);     // sqrt(0)==0 on diagonal
            if (row == col) val += lam;
        } else if (row < NK) {
            val = (col == NK) ? 1.0f : xs[row][col - NK - 1];
        } else if (col < NK) {
            val = (row == NK) ? 1.0f : xs[col][row - NK - 1];
        } else {
            val = 0.0f;
        }
        K[row][col] = val;
    }
    __syncthreads();

    // Gauss-Jordan with partial pivoting; physical row swaps so the final
    // augmented column is the solution in natural order.
    for (int p = 0; p < NF; ++p) {
        if (tid == 0) {
            int best = p;
            float bestAbs = fabsf(K[p][p]);
            for (int r = p + 1; r < NF; ++r) {
                const float a = fabsf(K[r][p]);
                if (a > bestAbs) { bestAbs = a; best = r; }
            }
            pivIdx = best;
        }
        __syncthreads();
        const int pr = pivIdx;
        if (pr != p) {
            for (int c = tid; c < NF + 1; c += bd) {
                const float t = K[p][c];
                K[p][c]  = K[pr][c];
                K[pr][c] = t;
            }
        }
        __syncthreads();
        float pv = K[p][p];
        if (fabsf(pv) < 1e-20f) pv = (pv < 0.0f) ? -1e-20f : 1e-20f;
        for (int c = tid; c < NF + 1; c += bd) piv[c] = K[p][c] / pv;
        __syncthreads();
        for (int c = tid; c < NF + 1; c += bd) K[p][c] = piv[c];
        if (tid < NF) fac[tid] = K[tid][p];
        __syncthreads();
        for (int idx = tid; idx < NF * (NF + 1); idx += bd) {
            const int r = idx / (NF + 1);
            const int c = idx % (NF + 1);
            if (r != p) K[r][c] -= fac[r] * piv[c];
        }
        __syncthreads();
    }

    float* W = wpad + b * (NF * 16);
    if (tid < NF) W[tid * 16 + ch] = K[tid][NF];
    if (ch == 0) {   // zero the 13 padding columns once per batch
        for (int idx = tid; idx < NF * 13; idx += bd) {
            const int r = idx / 13;
            const int c = 3 + idx % 13;
            W[r * 16 + c] = 0.0f;
        }
    }
}

// ---------------------------------------------------------------------------
// Kernel 2: streaming WMMA apply. Grid: BATCH*BLOCKS_PER_BATCH blocks,
// 256 threads (8 waves); each wave owns one 16-pixel tile.
// ---------------------------------------------------------------------------
__global__ void __launch_bounds__(256)
tps_apply_kernel(const float* __restrict__ raw,
                 const float* __restrict__ params,
                 const float* __restrict__ wpad,
                 float* __restrict__ out) {
    __shared__ float sW[NF * 16];          // padded weights (B operand source)
    __shared__ float sxs[NK][3];           // knots
    __shared__ float sOut[WAVES_PER_BLOCK][48];

    const int tid  = threadIdx.x;
    const int wave = tid >> 5;
    const int lane = tid & 31;
    const int b     = blockIdx.x / BLOCKS_PER_BATCH;
    const int tile0 = (blockIdx.x % BLOCKS_PER_BATCH) * WAVES_PER_BLOCK;

    // Stage weights and knots for this batch (shared by all 8 waves).
    for (int i = tid; i < NF * 16; i += 256) sW[i] = wpad[b * (NF * 16) + i];
    {
        const float* P = params + b * PARAM_STRIDE;
        if (tid < NK * 3) sxs[tid / 3][tid % 3] = P[tid];
    }
    __syncthreads();

    const int p0 = (tile0 + wave) * 16;     // first pixel of this wave's tile
    const int m  = lane & 15;               // pixel within tile (both halves)
    const int h  = lane >> 4;               // half-wave -> K-offset of 2

    // Pixel "rgb" = 3 consecutive floats (raw.reshape(B,HW,3) semantics).
    const float* px = raw + (size_t)b * PIX_PER_BATCH_ELEMS + (size_t)(p0 + m) * 3;
    const float r  = px[0];
    const float g  = px[1];
    const float bl = px[2];

    v8f acc = {};
    #pragma unroll
    for (int k = 0; k < 9; ++k) {
        const int f0 = 4 * k + 2 * h;       // features handled by this lane
        v2f a, bo;
        #pragma unroll
        for (int v = 0; v < 2; ++v) {
            const int f = f0 + v;
            float feat;
            if (f < NK) {
                const float dx = r  - sxs[f][0];
                const float dy = g  - sxs[f][1];
                const float dz = bl - sxs[f][2];
                const float sq = dx * dx + dy * dy + dz * dz;
                feat = __builtin_amdgcn_sqrtf(sq);   // raw v_sqrt_f32, <=1 ulp
            } else if (f == NK) {
                feat = 1.0f;
            } else {
                feat = (f == NK + 1) ? r : ((f == NK + 2) ? g : bl);
            }
            if (v == 0) a.x = feat; else a.y = feat;
        }
        bo.x = sW[(f0 + 0) * 16 + m];
        bo.y = sW[(f0 + 1) * 16 + m];
        // D(16x16,f32) += A(16x4,f32) x B(4x16,f32)
        acc = __builtin_amdgcn_wmma_f32_16x16x4_f32(
            /*neg_a=*/false, a, /*neg_b=*/false, bo,
            /*c_mod=*/(short)0, acc, /*reuse_a=*/false, /*reuse_b=*/false);
    }

    // D layout: vgpr rr, lane l -> M = rr + 8*(l>=16), N = l&15.
    // Stage valid channels (N<3) into LDS, pixel-major.
    if (m < 3) {
        #pragma unroll
        for (int rr = 0; rr < 8; ++rr) {
            sOut[wave][(rr + 8 * h) * 3 + m] = acc[rr];
        }
    }
    __syncthreads();

    // Coalesced store: 48 consecutive floats per wave (16 pixels * 3 ch).
    float* op = out + (size_t)b * PIX_PER_BATCH_ELEMS + (size_t)p0 * 3;
    for (int i = lane; i < 48; i += 32) op[i] = sOut[wave][i];
}

// ---------------------------------------------------------------------------
extern "C" void kernel_launch(void* const* d_in, const int* in_sizes, int n_in,
                              void* d_out, int out_size, void* d_ws, size_t ws_size,
                              hipStream_t stream) {
    (void)in_sizes; (void)n_in; (void)out_size; (void)ws_size;
    const float* raw    = (const float*)d_in[0];
    const float* params = (const float*)d_in[1];
    float* out  = (float*)d_out;
    float* wpad = (float*)d_ws;   // BATCH * 36 * 16 floats = 18,432 B

    tps_solve_kernel<<<BATCH * 3, 128, 0, stream>>>(params, wpad);
    tps_apply_kernel<<<BATCH * BLOCKS_PER_BATCH, 256, 0, stream>>>(raw, params, wpad, out);
}